// Propagation_prob_10067403341830
// MI455X (gfx1250) — compile-verified
//
#include <hip/hip_runtime.h>

// Problem shape (fixed by the reference): prob_volume [B=2, C=1, D=48, H=256, W=512] fp32
// Output: [B, 5, D, H, W] fp32 — five edge-clamped (dh,dw) shifts:
//   ch0:(-1,-1) ch1:(0,0) ch2:(+1,+1) ch3:(+1,-1) ch4:(-1,+1)
#define B_   2
#define D_   48
#define H_   256
#define W_   512
#define COUT 5

typedef float v4f __attribute__((ext_vector_type(4)));

__global__ __launch_bounds__(128)
void Propagation_prob_kernel(const float* __restrict__ in, float* __restrict__ out) {
    __shared__ float lds[3][W_];                 // rows h-1, h, h+1 (clamped), 6 KB

    const int t   = threadIdx.x;                 // 0..127, each owns one float4 of the row
    const int blk = blockIdx.x;                  // (b*D + d)*H + h
    const int h   = blk % H_;
    const int bd  = blk / H_;                    // b*D + d
    const int hm  = (h > 0)      ? h - 1 : 0;
    const int hp  = (h < H_ - 1) ? h + 1 : H_ - 1;

    const long inRowBase = (long)bd * (long)(H_ * W_);
    const int rows[3] = { hm, h, hp };

    // ---- CDNA5 async data path: stage 3 input rows into LDS, b128 per lane ----
    // Default (RT) temporal hint on loads: neighboring workgroups re-read these
    // rows ~3x, and the whole 50 MB input fits in the 192 MB L2.
    #pragma unroll
    for (int r = 0; r < 3; ++r) {
        const float* gsrc = in + inRowBase + (long)rows[r] * W_ + t * 4;
        unsigned long long gaddr = (unsigned long long)gsrc;            // VGPR pair
        unsigned ldsOff = (unsigned)(unsigned long long)&lds[r][t * 4]; // LDS address = low 32 bits of pointer
        asm volatile("global_load_async_to_lds_b128 %0, %1, off"
                     :: "v"(ldsOff), "v"(gaddr)
                     : "memory");
    }
    // Wait for this wave's async copies, then sync the workgroup (other waves' rows).
    asm volatile("s_wait_asynccnt 0" ::: "memory");
    __syncthreads();

    const int b  = bd / D_;
    const int d  = bd % D_;
    const int w0 = t * 4;

    const int dh[COUT] = { -1, 0, 1,  1, -1 };
    const int dw[COUT] = { -1, 0, 1, -1,  1 };

    #pragma unroll
    for (int c = 0; c < COUT; ++c) {
        const int   r  = dh[c] + 1;              // LDS row: 0=h-1, 1=h, 2=h+1
        const int   wa = w0 + dw[c];             // wa in [-1, 509]
        // Only the extreme lanes hit the W clamp; middle two reads are always in range.
        const float x0 = lds[r][wa < 0 ? 0 : wa];
        const float x1 = lds[r][wa + 1];
        const float x2 = lds[r][wa + 2];
        const int   w3 = wa + 3;
        const float x3 = lds[r][w3 > (W_ - 1) ? (W_ - 1) : w3];

        v4f v; v.x = x0; v.y = x1; v.z = x2; v.w = x3;
        const long outIdx = ((((long)b * COUT + c) * D_ + d) * (long)H_ + h) * W_ + w0;
        // Output is write-once streaming (252 MB) — non-temporal store hint keeps
        // the L2 free for the reused input rows instead of churning write data.
        __builtin_nontemporal_store(v, (v4f*)(out + outIdx));
    }
}

extern "C" void kernel_launch(void* const* d_in, const int* in_sizes, int n_in,
                              void* d_out, int out_size, void* d_ws, size_t ws_size,
                              hipStream_t stream) {
    (void)in_sizes; (void)n_in; (void)out_size; (void)d_ws; (void)ws_size;
    const float* in  = (const float*)d_in[0];
    float*       out = (float*)d_out;
    dim3 grid(B_ * D_ * H_);      // 24,576 workgroups, one output row each
    dim3 block(128);              // 4 waves (wave32)
    Propagation_prob_kernel<<<grid, block, 0, stream>>>(in, out);
}